// YOLOXHead_yzf_28552942584114
// MI455X (gfx1250) — compile-verified
//
#include <hip/hip_runtime.h>
#include <stdint.h>

#define NATT   85
#define STILE  128
#define SPAD   132      // 132 mod 64 banks = 4 -> only 2-way conflict on transposed reads
#define NROWS  25200

#if defined(__has_builtin)
#  if __has_builtin(__builtin_amdgcn_global_load_async_to_lds_b128)
#    define HAS_ASYNC_COPY 1
#  endif
#  if __has_builtin(__builtin_amdgcn_s_wait_asynccnt)
#    define HAS_WAIT_ASYNC 1
#  endif
#endif

typedef int v4i __attribute__((ext_vector_type(4)));
typedef __attribute__((address_space(1))) v4i* gbl_v4i_p;
typedef __attribute__((address_space(3))) v4i* lds_v4i_p;

__device__ __forceinline__ void wait_async_zero() {
#if defined(HAS_ASYNC_COPY)
#  if defined(HAS_WAIT_ASYNC)
    __builtin_amdgcn_s_wait_asynccnt(0);
#  else
    asm volatile("s_wait_asynccnt 0" ::: "memory");
#  endif
#endif
}

__device__ __forceinline__ float fast_sigmoid(float x) {
  // v_exp_f32 + v_add + v_rcp_f32 (avoid the IEEE-exact div expansion)
  return __builtin_amdgcn_rcpf(1.0f + __expf(-x));
}

// One block: (tile of up to 128 spatial positions) x (one anchor a) x (one batch n).
// Stage the 85x128 channel-major tile in LDS via async global->LDS DMA, then emit
// the fully-contiguous (spatial, attrib) output slab. Each of 255 active threads
// owns one attrib c: c>=4 threads apply sigmoid, c<4 threads decode their box
// coordinate directly from raw channels 0..3 (no separate decode pass needed).
__global__ __launch_bounds__(256) void yolox_decode_level(
    const float* __restrict__ pm, float* __restrict__ out,
    int W, int wShift /* log2(W/20) */, int HW, float stride,
    float wa0, float ha0, float wa1, float ha1, float wa2, float ha2,
    int lvlOff)
{
  __shared__ float lds[NATT * SPAD];

  const int tile = blockIdx.x;
  const int a    = blockIdx.y;
  const int n    = blockIdx.z;
  const int s0   = tile * STILE;
  const int validS = min(STILE, HW - s0);   // always a multiple of 16 here
  const int tid  = threadIdx.x;

  const size_t pmBase = ((size_t)n * (3 * NATT) + (size_t)a * NATT) * (size_t)HW + (size_t)s0;

  // ---- Phase 1: async-stage tile into LDS (coalesced along spatial) ----
  {
    const int vPerC  = validS >> 2;               // 16B chunks per channel (power of 2: 4/16/32)
    const int vShift = 31 - __clz(vPerC);
    const int vMask  = vPerC - 1;
    const int totalV = NATT * vPerC;
    for (int t = tid; t < totalV; t += 256) {
      const int c = t >> vShift;
      const int v = t & vMask;
      const float* g = pm + pmBase + (size_t)c * (size_t)HW + (v << 2);
      float*       l = &lds[c * SPAD + (v << 2)];
#if defined(HAS_ASYNC_COPY)
      __builtin_amdgcn_global_load_async_to_lds_b128(
          (gbl_v4i_p)g, (lds_v4i_p)l, 0, 0);
#else
      *(float4*)l = *(const float4*)g;
#endif
    }
    wait_async_zero();
  }
  __syncthreads();

  // ---- Phase 2: transposed LDS read, fused decode/sigmoid, contiguous NT store ----
  // 255 active threads = 3 spatial rows x 85 attribs; thread strides s += 3.
  // Per iteration the block covers t = tid + 255k: a contiguous output window.
  if (tid < 255) {
    const int sBase = (tid * 771) >> 16;         // tid/85 except at tid = 85, 170 (one less)
    const int c     = tid - sBase * 85;
    float* op = out +
        (((size_t)n * NROWS) + (size_t)lvlOff + (size_t)a * (size_t)HW + (size_t)s0) * NATT + c;

    if (c >= 4) {
      // objectness / class scores: sigmoid
      const float* lrow = &lds[c * SPAD];
      for (int s = sBase; s < validS; s += 3) {
        const float v = fast_sigmoid(lrow[s]);
        __builtin_nontemporal_store(v, op + (size_t)s * NATT);
      }
    } else {
      // box coords: out[c] = stride*(p_{c&1} + grid) +/- 0.5*exp(p_{2+(c&1)})*anchor
      const float wa = (a == 0) ? wa0 : ((a == 1) ? wa1 : wa2);
      const float ha = (a == 0) ? ha0 : ((a == 1) ? ha1 : ha2);
      const int   odd = c & 1;
      const float dim = odd ? ha : wa;
      const float sgn = (c >= 2) ? 0.5f : -0.5f;
      const float* lA = &lds[odd * SPAD];        // p0 or p1
      const float* lB = &lds[(2 + odd) * SPAD];  // p2 or p3
      for (int s = sBase; s < validS; s += 3) {
        const int g  = s0 + s;
        // exact g / W for W = 20 << wShift, g < 6400:  gy = ((g>>k)*3277)>>16
        const int gy = (int)(((unsigned)(g >> wShift) * 3277u) >> 16);
        const int gx = g - gy * W;
        const float coord = (float)(odd ? gy : gx);
        const float v = stride * (lA[s] + coord) + sgn * __expf(lB[s]) * dim;
        __builtin_nontemporal_store(v, op + (size_t)s * NATT);
      }
    }
  }
}

extern "C" void kernel_launch(void* const* d_in, const int* in_sizes, int n_in,
                              void* d_out, int out_size, void* d_ws, size_t ws_size,
                              hipStream_t stream) {
  const float* pm0 = (const float*)d_in[0];
  const float* pm1 = (const float*)d_in[1];
  const float* pm2 = (const float*)d_in[2];
  float* out = (float*)d_out;
  const int N = in_sizes[0] / (255 * 400);   // 32 for the reference shapes

  dim3 block(256);
  // Level 0: 20x20, stride 32, anchors (116,90)(156,198)(373,326), rows [0,1200)
  {
    const int HW = 400, tiles = (HW + STILE - 1) / STILE;
    yolox_decode_level<<<dim3(tiles, 3, N), block, 0, stream>>>(
        pm0, out, 20, 0, HW, 32.f, 116.f, 90.f, 156.f, 198.f, 373.f, 326.f, 0);
  }
  // Level 1: 40x40, stride 16, anchors (30,61)(62,45)(59,119), rows [1200,6000)
  {
    const int HW = 1600, tiles = (HW + STILE - 1) / STILE;
    yolox_decode_level<<<dim3(tiles, 3, N), block, 0, stream>>>(
        pm1, out, 40, 1, HW, 16.f, 30.f, 61.f, 62.f, 45.f, 59.f, 119.f, 1200);
  }
  // Level 2: 80x80, stride 8, anchors (10,13)(16,30)(33,23), rows [6000,25200)
  {
    const int HW = 6400, tiles = (HW + STILE - 1) / STILE;
    yolox_decode_level<<<dim3(tiles, 3, N), block, 0, stream>>>(
        pm2, out, 80, 2, HW, 8.f, 10.f, 13.f, 16.f, 30.f, 33.f, 23.f, 6000);
  }
}